// EmbAGNN_69157563400720
// MI455X (gfx1250) — compile-verified
//
#include <hip/hip_runtime.h>
#include <math.h>

typedef __attribute__((ext_vector_type(2))) float v2f;
typedef __attribute__((ext_vector_type(8))) float v8f;

#define NN     16384
#define NPE    4096        // nodes per event (4 events, contiguous)
#define KNN    30
#define NEDGE  (NN*KNN)    // 491520
#define EHID   32
#define RAD2   2.25f       // R^2 = 1.5^2

// ---------------- parameter bundles ----------------
struct InP   { const float *W,*b,*g,*bt; };                       // Linear -> LN -> tanh
struct MlpP  { const float *W0,*b0,*W1,*b1,*W2,*b2,*W3,*b3;
               const float *g0,*bt0,*g1,*bt1,*g2,*bt2; };         // 3x(Lin+LN+tanh) + head
struct EmbP  { const float *W0,*b0,*W1,*b1,*W2,*b2,*W3,*b3,*Wo,*bo; };

// ---------------- device helpers ----------------
__device__ __forceinline__ void ln_tanh8(float* h, const float* g, const float* bt) {
  float m = 0.f;
#pragma unroll
  for (int c = 0; c < 8; ++c) m += h[c];
  m *= 0.125f;
  float v = 0.f;
#pragma unroll
  for (int c = 0; c < 8; ++c) { float d = h[c] - m; v += d * d; }
  v *= 0.125f;
  float inv = rsqrtf(v + 1e-5f);
#pragma unroll
  for (int c = 0; c < 8; ++c) h[c] = tanhf((h[c] - m) * inv * g[c] + bt[c]);
}

template <int DIN>
__device__ __forceinline__ void lin8(const float* W, const float* b, const float* in, float* out) {
#pragma unroll
  for (int o = 0; o < 8; ++o) {
    float a = b[o];
#pragma unroll
    for (int i = 0; i < DIN; ++i) a += W[o * DIN + i] * in[i];
    out[o] = a;
  }
}

template <int DIN>
__device__ __forceinline__ void mlp3(const MlpP& p, const float* in, float* h) {
  float t[8];
  lin8<DIN>(p.W0, p.b0, in, h); ln_tanh8(h, p.g0, p.bt0);
  lin8<8>(p.W1, p.b1, h, t);    ln_tanh8(t, p.g1, p.bt1);
  lin8<8>(p.W2, p.b2, t, h);    ln_tanh8(h, p.g2, p.bt2);
}

__device__ __forceinline__ void input_net(const InP& p, const float* x3, float* out8) {
  lin8<3>(p.W, p.b, x3, out8);
  ln_tanh8(out8, p.g, p.bt);
}

__device__ __forceinline__ void emb_fwd(const EmbP& p, const float* in19, float* out8) {
  float h[EHID], t[EHID];
  for (int o = 0; o < EHID; ++o) { float a = p.b0[o];
#pragma unroll
    for (int i = 0; i < 19; ++i) a += p.W0[o * 19 + i] * in19[i]; h[o] = tanhf(a); }
  for (int o = 0; o < EHID; ++o) { float a = p.b1[o];
    for (int i = 0; i < EHID; ++i) a += p.W1[o * EHID + i] * h[i]; t[o] = tanhf(a); }
  for (int o = 0; o < EHID; ++o) { float a = p.b2[o];
    for (int i = 0; i < EHID; ++i) a += p.W2[o * EHID + i] * t[i]; h[o] = tanhf(a); }
  for (int o = 0; o < EHID; ++o) { float a = p.b3[o];
    for (int i = 0; i < EHID; ++i) a += p.W3[o * EHID + i] * h[i]; t[o] = tanhf(a); }
#pragma unroll
  for (int o = 0; o < 8; ++o) { float a = p.bo[o];
    for (int i = 0; i < EHID; ++i) a += p.Wo[o * EHID + i] * t[i]; out8[o] = a; }
}

// ---------------- kernels ----------------

// 1) initial feature net + spatial net + embedding net (one thread / node)
__global__ __launch_bounds__(256)
void k_init_nodes(const float* __restrict__ x, InP featp, InP spatp, EmbP embp,
                  float* __restrict__ feat, float* __restrict__ spatial, float* __restrict__ sq) {
  int i = blockIdx.x * blockDim.x + threadIdx.x;
  if (i >= NN) return;
  float xi[3] = { x[i*3+0], x[i*3+1], x[i*3+2] };
  float f8[8], s8[8], in19[19], out8[8];
  input_net(featp, xi, f8);
  input_net(spatp, xi, s8);
#pragma unroll
  for (int c = 0; c < 3; ++c) in19[c]     = xi[c];
#pragma unroll
  for (int c = 0; c < 8; ++c) in19[3 + c]  = f8[c];
#pragma unroll
  for (int c = 0; c < 8; ++c) in19[11 + c] = s8[c];
  emb_fwd(embp, in19, out8);
  float s = 0.f;
#pragma unroll
  for (int c = 0; c < 8; ++c) { feat[i*8+c] = f8[c]; spatial[i*8+c] = out8[c]; s += out8[c]*out8[c]; }
  sq[i] = s;
}

// 2) radius graph: FP32 WMMA Gram tiles + async-to-LDS double-buffered B staging
//    + per-row top-30 selection split across all 32 lanes.
//    One wave per 16-row stripe; event-local 4096 columns in 256 16-col tiles.
__global__ __launch_bounds__(32)
void k_radius_wmma(const float* __restrict__ S, const float* __restrict__ sq,
                   int* __restrict__ startIdx, float* __restrict__ maskE) {
  const int stripe   = blockIdx.x;           // 0..1023
  const int rowBase  = stripe * 16;
  const int ev       = rowBase / NPE;
  const int colBase0 = ev * NPE;
  const int lane = threadIdx.x;              // 0..31
  const int m    = lane & 15;
  const int kh   = lane >> 4;                // K-half select (ISA 32-bit A/B layout)

  // A fragments: 16x4 f32 tiles, lane m = row M, VGPR0 = K {0|2}, VGPR1 = K {1|3}
  const int arow = rowBase + m;
  v2f aLo = *(const v2f*)(S + arow * 8 + 2 * kh);       // K 0..3
  v2f aHi = *(const v2f*)(S + arow * 8 + 4 + 2 * kh);   // K 4..7

  __shared__ float bbuf[2][16][8];          // double-buffered async B tiles
  __shared__ float tile[16][17];            // 16x16 distance tile
  __shared__ float bestD[16][2 * KNN];      // two half-lists per row
  __shared__ int   bestI[16][2 * KNN];

  // C/D layout: lane holds col N = lane&15; rows M = kh*8 + v
  float sqRow[8];
#pragma unroll
  for (int v = 0; v < 8; ++v) sqRow[v] = sq[rowBase + kh * 8 + v];

  // async-stage tile 0: each lane copies 16B (row m, half kh) -> ASYNCcnt path
  {
    const float* gp = S + (size_t)(colBase0 + m) * 8 + kh * 4;
    unsigned lp = (unsigned)(uintptr_t)&bbuf[0][m][kh * 4];
    asm volatile("global_load_async_to_lds_b128 %0, %1, off"
                 :: "v"(lp), "v"(gp) : "memory");
  }

  // per-lane selection state: lane handles row r, column sub-range nb..nb+7,
  // half-list slot base lbase. Each half scans 2048 candidates so both lists fill;
  // the union of the two 30-smallest half-lists contains the global 30-smallest.
  const int r     = lane & 15;
  const int nb    = (lane >> 4) * 8;
  const int lbase = (lane >> 4) * KNN;
  int   cnt = 0;
  float mxv = 1e38f;
  int   mxa = 0;

  for (int t = 0; t < NPE / 16; ++t) {
    const int colBase = colBase0 + t * 16;

    if (t + 1 < NPE / 16) {
      // prefetch next B tile into the other buffer, then wait for current (<=1 pending)
      const float* gp = S + (size_t)(colBase + 16 + m) * 8 + kh * 4;
      unsigned lp = (unsigned)(uintptr_t)&bbuf[(t + 1) & 1][m][kh * 4];
      asm volatile("global_load_async_to_lds_b128 %0, %1, off"
                   :: "v"(lp), "v"(gp) : "memory");
      asm volatile("s_wait_asynccnt 0x1" ::: "memory");
    } else {
      asm volatile("s_wait_asynccnt 0x0" ::: "memory");
    }

    // B fragments from LDS: lane holds col = colBase+m, K-half kh
    v2f bLo = *(const v2f*)&bbuf[t & 1][m][2 * kh];
    v2f bHi = *(const v2f*)&bbuf[t & 1][m][4 + 2 * kh];

    v8f acc = {};
    acc = __builtin_amdgcn_wmma_f32_16x16x4_f32(false, aLo, false, bLo, (short)0, acc, false, false);
    acc = __builtin_amdgcn_wmma_f32_16x16x4_f32(false, aHi, false, bHi, (short)0, acc, false, false);

    const float sqc = sq[colBase + m];
#pragma unroll
    for (int v = 0; v < 8; ++v) {
      const int M = kh * 8 + v;
      float d2 = sqRow[v] + sqc - 2.0f * acc[v];
      if (rowBase + M == colBase + m) d2 = 1e30f;      // no self loops
      tile[M][m] = d2;
    }
    __syncthreads();

    // all 32 lanes select: 8 candidates each (replace-max into own half-list)
#pragma unroll 2
    for (int n = 0; n < 8; ++n) {
      const float d = tile[r][nb + n];
      const int   c = colBase + nb + n;
      if (cnt < KNN) {
        bestD[r][lbase + cnt] = d; bestI[r][lbase + cnt] = c; ++cnt;
        if (cnt == KNN) {
          mxv = -1e38f;
          for (int j = 0; j < KNN; ++j)
            if (bestD[r][lbase + j] > mxv) { mxv = bestD[r][lbase + j]; mxa = j; }
        }
      } else if (d < mxv) {
        bestD[r][lbase + mxa] = d; bestI[r][lbase + mxa] = c;
        mxv = -1e38f;
        for (int j = 0; j < KNN; ++j)
          if (bestD[r][lbase + j] > mxv) { mxv = bestD[r][lbase + j]; mxa = j; }
      }
    }
    __syncthreads();
  }

  // merge the two half-lists: sort 60 ascending, emit first 30 (top_k order)
  if (lane < 16) {
    for (int i = 1; i < 2 * KNN; ++i) {
      float d = bestD[r][i]; int c = bestI[r][i]; int j = i - 1;
      while (j >= 0 && bestD[r][j] > d) { bestD[r][j+1] = bestD[r][j]; bestI[r][j+1] = bestI[r][j]; --j; }
      bestD[r][j+1] = d; bestI[r][j+1] = c;
    }
    const int node = rowBase + r;
    for (int k = 0; k < KNN; ++k) {
      startIdx[node * KNN + k] = bestI[r][k];
      maskE[node * KNN + k]    = (bestD[r][k] <= RAD2) ? 1.0f : 0.0f;
    }
  }
}

// 3) edge MLP: mode 0 -> e = sigmoid(logit)*mask ; mode 1 -> out = logit*mask
__global__ __launch_bounds__(256)
void k_edge(const float* __restrict__ feat, const int* __restrict__ startIdx,
            const float* __restrict__ maskE, MlpP p, float* __restrict__ out, int mode) {
  int e = blockIdx.x * blockDim.x + threadIdx.x;
  if (e >= NEDGE) return;
  const int tgt = e / KNN;               // end node
  const int src = startIdx[e];           // start node
  float in16[16];
#pragma unroll
  for (int c = 0; c < 8; ++c) { in16[c] = feat[src*8+c]; in16[8+c] = feat[tgt*8+c]; }
  float h[8];
  mlp3<16>(p, in16, h);
  float logit = p.b3[0];
#pragma unroll
  for (int c = 0; c < 8; ++c) logit += p.W3[c] * h[c];
  const float mk = maskE[e];
  out[e] = mode ? (logit * mk) : (mk / (1.0f + expf(-logit)));
}

// 4) mi = segment_sum(e*feat[start], end) — edges grouped by target: no atomics.
//    Also zero mo for this node (consumed by the scatter kernel that follows).
__global__ __launch_bounds__(256)
void k_agg_mi(const float* __restrict__ feat, const float* __restrict__ eVal,
              const int* __restrict__ startIdx, float* __restrict__ mi, float* __restrict__ mo) {
  int i = blockIdx.x * blockDim.x + threadIdx.x;
  if (i >= NN) return;
  float acc[8] = {0,0,0,0,0,0,0,0};
  for (int k = 0; k < KNN; ++k) {
    const float ev = eVal[i * KNN + k];
    const int   s  = startIdx[i * KNN + k];
#pragma unroll
    for (int c = 0; c < 8; ++c) acc[c] += ev * feat[s*8+c];
  }
#pragma unroll
  for (int c = 0; c < 8; ++c) { mi[i*8+c] = acc[c]; mo[i*8+c] = 0.f; }
}

// 5) mo = segment_sum(e*feat[end], start) — scatter with float atomics.
__global__ __launch_bounds__(256)
void k_scatter_mo(const float* __restrict__ feat, const float* __restrict__ eVal,
                  const int* __restrict__ startIdx, float* __restrict__ mo) {
  int e = blockIdx.x * blockDim.x + threadIdx.x;
  if (e >= NEDGE) return;
  const float ev = eVal[e];
  if (ev == 0.f) return;
  const int tgt = e / KNN;
  const int s   = startIdx[e];
#pragma unroll
  for (int c = 0; c < 8; ++c) atomicAdd(&mo[s*8+c], ev * feat[tgt*8+c]);
}

// 6) node MLP: fnew = node_mlp(cat(mi,mo,f0)); fres = f0 + fnew (next-iter features)
__global__ __launch_bounds__(256)
void k_node(const float* __restrict__ mi, const float* __restrict__ mo,
            const float* __restrict__ f0, MlpP p,
            float* __restrict__ fnew, float* __restrict__ fres) {
  int i = blockIdx.x * blockDim.x + threadIdx.x;
  if (i >= NN) return;
  float in24[24];
#pragma unroll
  for (int c = 0; c < 8; ++c) { in24[c] = mi[i*8+c]; in24[8+c] = mo[i*8+c]; in24[16+c] = f0[i*8+c]; }
  float h[8], o[8];
  mlp3<24>(p, in24, h);
  lin8<8>(p.W3, p.b3, h, o);
#pragma unroll
  for (int c = 0; c < 8; ++c) { fnew[i*8+c] = o[c]; fres[i*8+c] = f0[i*8+c] + o[c]; }
}

// 7) spatialNew = emb(cat(x, fnew, spatialOld)); also |spatialNew|^2
__global__ __launch_bounds__(256)
void k_emb(const float* __restrict__ x, const float* __restrict__ fnew,
           const float* __restrict__ spOld, EmbP p,
           float* __restrict__ spNew, float* __restrict__ sq) {
  int i = blockIdx.x * blockDim.x + threadIdx.x;
  if (i >= NN) return;
  float in19[19], out8[8];
#pragma unroll
  for (int c = 0; c < 3; ++c) in19[c]      = x[i*3+c];
#pragma unroll
  for (int c = 0; c < 8; ++c) in19[3 + c]  = fnew[i*8+c];
#pragma unroll
  for (int c = 0; c < 8; ++c) in19[11 + c] = spOld[i*8+c];
  emb_fwd(p, in19, out8);
  float s = 0.f;
#pragma unroll
  for (int c = 0; c < 8; ++c) { spNew[i*8+c] = out8[c]; s += out8[c]*out8[c]; }
  sq[i] = s;
}

// 8) pack tuple outputs: [logits | spatial | start | end] (indices as float values)
__global__ __launch_bounds__(256)
void k_finalize(const float* __restrict__ sp, const int* __restrict__ startIdx,
                float* __restrict__ out) {
  int e = blockIdx.x * blockDim.x + threadIdx.x;
  if (e >= NEDGE) return;
  if (e < NN * 8) out[NEDGE + e] = sp[e];
  out[NEDGE + NN*8 + e]         = (float)startIdx[e];
  out[NEDGE + NN*8 + NEDGE + e] = (float)(e / KNN);
}

// ---------------- host launch ----------------
extern "C" void kernel_launch(void* const* d_in, const int* in_sizes, int n_in,
                              void* d_out, int out_size, void* d_ws, size_t ws_size,
                              hipStream_t stream) {
  auto F = [&](int i) { return (const float*)d_in[i]; };
  const float* x = F(0);
  // d_in[1] = batch (events are contiguous 4096-blocks; derived by index)

  // pytree leaf order (dicts sorted by key: W<b, bt<g, lin<ln, lins<out)
  MlpP edgep;
  edgep.W0 = F(2);  edgep.b0 = F(3);  edgep.W1 = F(4);  edgep.b1 = F(5);
  edgep.W2 = F(6);  edgep.b2 = F(7);  edgep.W3 = F(8);  edgep.b3 = F(9);
  edgep.bt0 = F(10); edgep.g0 = F(11); edgep.bt1 = F(12); edgep.g1 = F(13);
  edgep.bt2 = F(14); edgep.g2 = F(15);
  EmbP embp;
  embp.W0 = F(16); embp.b0 = F(17); embp.W1 = F(18); embp.b1 = F(19);
  embp.W2 = F(20); embp.b2 = F(21); embp.W3 = F(22); embp.b3 = F(23);
  embp.Wo = F(24); embp.bo = F(25);
  InP featp; featp.W = F(26); featp.b = F(27); featp.bt = F(28); featp.g = F(29);
  MlpP nodep;
  nodep.W0 = F(30); nodep.b0 = F(31); nodep.W1 = F(32); nodep.b1 = F(33);
  nodep.W2 = F(34); nodep.b2 = F(35); nodep.W3 = F(36); nodep.b3 = F(37);
  nodep.bt0 = F(38); nodep.g0 = F(39); nodep.bt1 = F(40); nodep.g1 = F(41);
  nodep.bt2 = F(42); nodep.g2 = F(43);
  InP spatp; spatp.W = F(44); spatp.b = F(45); spatp.bt = F(46); spatp.g = F(47);

  // workspace carve-up (~9.6 MB)
  float* ws = (float*)d_ws;
  float* spA   = ws;              // N*8
  float* spB   = spA  + NN * 8;
  float* featA = spB  + NN * 8;
  float* featB = featA + NN * 8;
  float* fnew  = featB + NN * 8;
  float* sq    = fnew + NN * 8;   // N
  float* maskE = sq   + NN;       // NEDGE
  float* eVal  = maskE + NEDGE;
  float* mi    = eVal + NEDGE;    // N*8
  float* mo    = mi   + NN * 8;
  int*   startIdx = (int*)(mo + NN * 8);

  float* spCur = spA;  float* spNext = spB;
  float* fCur  = featA; float* fNext = featB;
  float* outF  = (float*)d_out;

  const int NB_NODE = (NN + 255) / 256;
  const int NB_EDGE = (NEDGE + 255) / 256;

  k_init_nodes<<<NB_NODE, 256, 0, stream>>>(x, featp, spatp, embp, fCur, spCur, sq);
  k_radius_wmma<<<NN / 16, 32, 0, stream>>>(spCur, sq, startIdx, maskE);

  for (int it = 0; it < 3; ++it) {
    k_edge<<<NB_EDGE, 256, 0, stream>>>(fCur, startIdx, maskE, edgep, eVal, 0);
    k_agg_mi<<<NB_NODE, 256, 0, stream>>>(fCur, eVal, startIdx, mi, mo);
    k_scatter_mo<<<NB_EDGE, 256, 0, stream>>>(fCur, eVal, startIdx, mo);
    k_node<<<NB_NODE, 256, 0, stream>>>(mi, mo, fCur, nodep, fnew, fNext);
    k_emb<<<NB_NODE, 256, 0, stream>>>(x, fnew, spCur, embp, spNext, sq);
    k_radius_wmma<<<NN / 16, 32, 0, stream>>>(spNext, sq, startIdx, maskE);
    { float* t = spCur; spCur = spNext; spNext = t; }
    { float* t = fCur;  fCur  = fNext;  fNext  = t; }
  }

  k_edge<<<NB_EDGE, 256, 0, stream>>>(fCur, startIdx, maskE, edgep, outF, 1);
  k_finalize<<<NB_EDGE, 256, 0, stream>>>(spCur, startIdx, outF);
  (void)in_sizes; (void)n_in; (void)out_size; (void)ws_size;
}